// EdgeAwareGATLayer_32238024524456
// MI455X (gfx1250) — compile-verified
//
#include <hip/hip_runtime.h>

typedef float v2f __attribute__((ext_vector_type(2)));
typedef float v8f __attribute__((ext_vector_type(8)));

#define NIN   64
#define EDIM  32
#define KDIM  96      // NIN + EDIM
#define NOUT  128
#define NHEAD 4
#define ATTK  160     // 2*NIN + EDIM

#define BE       128  // edges per workgroup in the WMMA kernel
#define ASTRIDE  98   // LDS row stride (floats) for msg_in tile (96 + pad, even, conflict-free)
#define WPSTRIDE 144  // LDS row stride (float2) for K-paired W^T: 1152B/row -> halves on disjoint banks
#define NPAIR    48   // KDIM/2 K-pairs

// ---------------------------------------------------------------------------
// Kernel 0: init output accumulator + softmax segment buffers
// ---------------------------------------------------------------------------
__global__ __launch_bounds__(256) void k_init(float* __restrict__ out,
                                              float* __restrict__ segmax,
                                              float* __restrict__ segsum,
                                              int n_out, int n_seg) {
  int i = blockIdx.x * 256 + threadIdx.x;
  if (i < n_out) out[i] = 0.0f;
  if (i < n_seg) { segmax[i] = -3.0e38f; segsum[i] = 0.0f; }
}

// ---------------------------------------------------------------------------
// Kernel 1: per-edge attention logits + LeakyReLU + segment max (float atomic)
// ---------------------------------------------------------------------------
__global__ __launch_bounds__(256) void k_logits(const float* __restrict__ x,
                                                const int* __restrict__ src,
                                                const int* __restrict__ dst,
                                                const float* __restrict__ edge_attr,
                                                const float* __restrict__ W_att,
                                                float* __restrict__ logits,
                                                float* __restrict__ segmax,
                                                int E) {
  __shared__ float sW[NHEAD * ATTK];
  for (int i = threadIdx.x; i < NHEAD * ATTK; i += 256) sW[i] = W_att[i];
  __syncthreads();
  int e = blockIdx.x * 256 + threadIdx.x;
  if (e >= E) return;
  int s = src[e], d = dst[e];
  const float* xs = x + (long long)s * NIN;
  const float* xd = x + (long long)d * NIN;
  const float* ea = edge_attr + (long long)e * EDIM;
  float a0 = 0.f, a1 = 0.f, a2 = 0.f, a3 = 0.f;
#pragma unroll 8
  for (int k = 0; k < NIN; ++k) {
    float v = xs[k];
    a0 += v * sW[k]; a1 += v * sW[ATTK + k];
    a2 += v * sW[2 * ATTK + k]; a3 += v * sW[3 * ATTK + k];
  }
#pragma unroll 8
  for (int k = 0; k < NIN; ++k) {
    float v = xd[k];
    a0 += v * sW[NIN + k]; a1 += v * sW[ATTK + NIN + k];
    a2 += v * sW[2 * ATTK + NIN + k]; a3 += v * sW[3 * ATTK + NIN + k];
  }
#pragma unroll 8
  for (int k = 0; k < EDIM; ++k) {
    float v = ea[k];
    a0 += v * sW[2 * NIN + k]; a1 += v * sW[ATTK + 2 * NIN + k];
    a2 += v * sW[2 * ATTK + 2 * NIN + k]; a3 += v * sW[3 * ATTK + 2 * NIN + k];
  }
  float l[NHEAD] = {a0, a1, a2, a3};
  long long base = (long long)e * NHEAD;
#pragma unroll
  for (int h = 0; h < NHEAD; ++h) {
    float v = l[h];
    v = (v > 0.f) ? v : 0.2f * v;            // LeakyReLU(0.2)
    logits[base + h] = v;
    __hip_atomic_fetch_max(&segmax[(long long)d * NHEAD + h], v,
                           __ATOMIC_RELAXED, __HIP_MEMORY_SCOPE_AGENT);
  }
}

// ---------------------------------------------------------------------------
// Kernel 2: exp(l - segmax[dst]) in-place + segment sum
// ---------------------------------------------------------------------------
__global__ __launch_bounds__(256) void k_expsum(const int* __restrict__ dst,
                                                float* __restrict__ logits,
                                                const float* __restrict__ segmax,
                                                float* __restrict__ segsum,
                                                long long total) {
  long long i = (long long)blockIdx.x * 256 + threadIdx.x;
  if (i >= total) return;
  int h = (int)(i & 3);
  long long e = i >> 2;
  int d = dst[e];
  float m = segmax[(long long)d * NHEAD + h];
  float ex = expf(logits[i] - m);
  logits[i] = ex;                            // buffer now holds exp_l
  atomicAdd(&segsum[(long long)d * NHEAD + h], ex);
}

// ---------------------------------------------------------------------------
// Kernel 3: message GEMM via V_WMMA_F32_16X16X4_F32 + attention-weighted
//           scatter-add into out accumulator (d_out).
// 256 threads = 8 waves; wave w owns edges [16w, 16w+16) of a 128-edge tile
// and computes all 8 column tiles (128 outputs).
// Gather uses GLOBAL_LOAD_ASYNC_TO_LDS_B128 (cache -> LDS, ASYNCcnt).
// ---------------------------------------------------------------------------
__global__ __launch_bounds__(256) void k_msg_wmma(const float* __restrict__ x,
                                                  const int* __restrict__ src,
                                                  const int* __restrict__ dst,
                                                  const float* __restrict__ edge_attr,
                                                  const float* __restrict__ W_msg,
                                                  const float* __restrict__ expl,
                                                  const float* __restrict__ segsum,
                                                  float* __restrict__ out,
                                                  int E) {
  __shared__ float sA[BE * ASTRIDE];         // 128 x 96 msg_in tile (padded)
  __shared__ v2f   sWp[NPAIR * WPSTRIDE];    // K-paired W^T: sWp[k/2][n] = {W^T[k][n], W^T[k+1][n]}
  __shared__ float sAtt[BE * NHEAD];         // softmax weights per (edge, head)
  __shared__ int   sSrc[BE];
  __shared__ int   sDst[BE];

  const int tid = threadIdx.x;
  const long long ebase = (long long)blockIdx.x * BE;

  // Stage edge ids + normalized attention weights first (gather needs sSrc).
  for (int i = tid; i < BE; i += 256) {
    long long ge = ebase + i;
    int s = 0, d = 0;
    if (ge < E) { s = src[ge]; d = dst[ge]; }
    sSrc[i] = s;
    sDst[i] = d;
#pragma unroll
    for (int h = 0; h < NHEAD; ++h) {
      float a = 0.f;
      if (ge < E)
        a = expl[ge * NHEAD + h] / (segsum[(long long)d * NHEAD + h] + 1e-9f);
      sAtt[i * NHEAD + h] = a;
    }
  }
  __syncthreads();

  // Async gather [x[src] ; edge_attr] directly into LDS (no VGPR round-trip).
  // 128 edges x 24 float4 = 12 async B128 copies per thread.
  for (int i = tid; i < BE * 24; i += 256) {
    int e = i / 24, q = i % 24;
    long long ge = ebase + e;
    if (ge < E) {
      const float* gsrc = (q < 16)
          ? (x + (long long)sSrc[e] * NIN + q * 4)
          : (edge_attr + ge * EDIM + (q - 16) * 4);
      unsigned lds_off = (unsigned)(uintptr_t)(&sA[e * ASTRIDE + q * 4]);
      unsigned long long ga = (unsigned long long)(uintptr_t)gsrc;
      asm volatile("global_load_async_to_lds_b128 %0, %1, off"
                   :: "v"(lds_off), "v"(ga) : "memory");
    }
  }

  // Stage W_msg transposed + K-paired while the async copies are in flight.
  {
    float* sWf = (float*)sWp;
    for (int i = tid; i < NOUT * KDIM; i += 256) {
      int k = i % KDIM, n = i / KDIM;        // W_msg is (NOUT, KDIM) row-major
      sWf[((k >> 1) * WPSTRIDE + n) * 2 + (k & 1)] = W_msg[i];
    }
  }

  asm volatile("s_wait_asynccnt 0x0" ::: "memory");
  __syncthreads();

  const int wave  = tid >> 5;                // 0..7 : edge tile
  const int lane  = tid & 31;
  const int half  = lane >> 4;               // K-half selector for A/B frags
  const int l16   = lane & 15;               // M row (A) / N col (B)
  const int erow0 = wave * 16;

  v8f acc[8];
#pragma unroll
  for (int n = 0; n < 8; ++n) acc[n] = 0.0f;

  for (int k0 = 0; k0 < KDIM; k0 += 4) {
    // A fragment: lanes 0-15: {A[l][k0], A[l][k0+1]}; lanes 16-31: k0+2, k0+3.
    v2f a = *(const v2f*)&sA[(erow0 + l16) * ASTRIDE + k0 + 2 * half];
    int p = (k0 >> 1) + half;                // K-pair index for B fragment
#pragma unroll
    for (int n = 0; n < 8; ++n) {
      v2f b = sWp[p * WPSTRIDE + n * 16 + l16];   // single ds_load_b64
      acc[n] = __builtin_amdgcn_wmma_f32_16x16x4_f32(
          false, a, false, b, (short)0, acc[n], false, false);
    }
  }

  // Scale by attention and scatter-add. D layout: acc[n][v] is
  // row M = v + 8*half of the 16-edge tile, column n*16 + l16.
#pragma unroll
  for (int n = 0; n < 8; ++n) {
    int col  = n * 16 + l16;
    int head = n >> 1;                       // 32 cols per head
#pragma unroll
    for (int v = 0; v < 8; ++v) {
      int elocal = erow0 + v + 8 * half;
      long long ge = ebase + elocal;
      if (ge < E) {
        float w = acc[n][v] * sAtt[elocal * NHEAD + head];
        atomicAdd(&out[(long long)sDst[elocal] * NOUT + col], w);
      }
    }
  }
}

// ---------------------------------------------------------------------------
// Kernel 4: bias + LayerNorm, in place on d_out. One 128-thread block per row.
// ---------------------------------------------------------------------------
__global__ __launch_bounds__(128) void k_layernorm(float* __restrict__ out,
                                                   const float* __restrict__ bias,
                                                   const float* __restrict__ gamma,
                                                   const float* __restrict__ beta,
                                                   int N) {
  __shared__ float red[NOUT];
  int row = blockIdx.x, t = threadIdx.x;
  if (row >= N) return;
  float v = out[(long long)row * NOUT + t] + bias[t];
  red[t] = v;
  __syncthreads();
  for (int s = 64; s > 0; s >>= 1) {
    if (t < s) red[t] += red[t + s];
    __syncthreads();
  }
  float mu = red[0] * (1.0f / NOUT);
  __syncthreads();
  float dv = v - mu;
  red[t] = dv * dv;
  __syncthreads();
  for (int s = 64; s > 0; s >>= 1) {
    if (t < s) red[t] += red[t + s];
    __syncthreads();
  }
  float var = red[0] * (1.0f / NOUT);
  out[(long long)row * NOUT + t] = dv * rsqrtf(var + 1e-5f) * gamma[t] + beta[t];
}

// ---------------------------------------------------------------------------
extern "C" void kernel_launch(void* const* d_in, const int* in_sizes, int n_in,
                              void* d_out, int out_size, void* d_ws, size_t ws_size,
                              hipStream_t stream) {
  const float* x         = (const float*)d_in[0];
  const int*   eidx      = (const int*)d_in[1];   // (2, E) int32
  const float* edge_attr = (const float*)d_in[2];
  const float* W_msg     = (const float*)d_in[3];
  const float* W_att     = (const float*)d_in[4];
  const float* bias      = (const float*)d_in[5];
  const float* gamma     = (const float*)d_in[6];
  const float* beta      = (const float*)d_in[7];

  const int N = in_sizes[0] / NIN;
  const int E = in_sizes[1] / 2;
  const int* src = eidx;
  const int* dst = eidx + E;

  float* out = (float*)d_out;
  char*  ws  = (char*)d_ws;
  float* segmax = (float*)ws;                                   // N*4 f32
  float* segsum = (float*)(ws + (size_t)N * NHEAD * 4);         // N*4 f32
  float* logits = (float*)(ws + (size_t)N * NHEAD * 8);         // E*4 f32 (-> exp_l)

  const int n_out = N * NOUT;
  const int n_seg = N * NHEAD;

  k_init<<<(n_out + 255) / 256, 256, 0, stream>>>(out, segmax, segsum, n_out, n_seg);

  k_logits<<<(E + 255) / 256, 256, 0, stream>>>(x, src, dst, edge_attr, W_att,
                                                logits, segmax, E);

  long long total = (long long)E * NHEAD;
  k_expsum<<<(int)((total + 255) / 256), 256, 0, stream>>>(dst, logits, segmax,
                                                           segsum, total);

  k_msg_wmma<<<(E + BE - 1) / BE, 256, 0, stream>>>(x, src, dst, edge_attr, W_msg,
                                                    logits, segsum, out, E);

  k_layernorm<<<N, 128, 0, stream>>>(out, bias, gamma, beta, N);
}